// DirPCAPassConv_81329500717451
// MI455X (gfx1250) — compile-verified
//
#include <hip/hip_runtime.h>
#include <hip/hip_bf16.h>

typedef __attribute__((ext_vector_type(2))) float v2f;
typedef __attribute__((ext_vector_type(4))) float v4f;
typedef __attribute__((ext_vector_type(8))) float v8f;

#define FEAT_D 128
#define HID_D  384   // 3 * FEAT_D

// ---------------------------------------------------------------------------
// Zero workspace (sums + degrees), b128 stores + scalar tail
// ---------------------------------------------------------------------------
__global__ __launch_bounds__(256) void zero_ws_kernel(float* ws, size_t count) {
    size_t nvec = count >> 2;
    v4f* ws4 = (v4f*)ws;
    size_t i = (size_t)blockIdx.x * blockDim.x + threadIdx.x;
    size_t stride = (size_t)gridDim.x * blockDim.x;
    v4f z = {};
    for (; i < nvec; i += stride) ws4[i] = z;
    // tail
    size_t t = (nvec << 2) + ((size_t)blockIdx.x * blockDim.x + threadIdx.x);
    if (t < count) ws[t] = 0.0f;
}

// ---------------------------------------------------------------------------
// Edge scatter: one WAVE per edge, one thread per 4 channels (b128 gathers).
//   sum_fwd[dst] += feat[src]; sum_bwd[src] += feat[dst]
//   lane 0 also bumps the float degree counters (fused degree kernel).
// ---------------------------------------------------------------------------
__global__ __launch_bounds__(256) void edge_scatter_kernel(const int* __restrict__ src,
                                                           const int* __restrict__ dst,
                                                           const float* __restrict__ feat,
                                                           float* __restrict__ sum_fwd,
                                                           float* __restrict__ sum_bwd,
                                                           float* __restrict__ deg_fwd,
                                                           float* __restrict__ deg_bwd,
                                                           int n_edges) {
    long long t = (long long)blockIdx.x * blockDim.x + threadIdx.x;
    int e    = (int)(t >> 5);        // edge id (uniform per wave)
    int lane = (int)(t & 31);        // lane -> 4-channel chunk
    if (e >= n_edges) return;

    int s = src[e];
    int d = dst[e];

    const int c = lane << 2;         // channel base, 0..124
    v4f fs = *(const v4f*)(feat + (size_t)s * FEAT_D + c);   // global_load_b128
    v4f fd = *(const v4f*)(feat + (size_t)d * FEAT_D + c);

    float* pf = sum_fwd + (size_t)d * FEAT_D + c;
    float* pb = sum_bwd + (size_t)s * FEAT_D + c;
#pragma unroll
    for (int j = 0; j < 4; ++j) {
        atomicAdd(pf + j, fs[j]);
        atomicAdd(pb + j, fd[j]);
    }
    if (lane == 0) {
        atomicAdd(deg_fwd + d, 1.0f);
        atomicAdd(deg_bwd + s, 1.0f);
    }
}

// ---------------------------------------------------------------------------
// Finalize: mean = sum / max(deg, 1). One wave per node, b128 ldst.
// ---------------------------------------------------------------------------
__global__ __launch_bounds__(256) void finalize_kernel(float* __restrict__ sum_fwd,
                                                       float* __restrict__ sum_bwd,
                                                       const float* __restrict__ deg_fwd,
                                                       const float* __restrict__ deg_bwd,
                                                       int n_nodes) {
    long long t = (long long)blockIdx.x * blockDim.x + threadIdx.x;
    int n    = (int)(t >> 5);
    int lane = (int)(t & 31);
    if (n >= n_nodes) return;

    float inv_f = 1.0f / fmaxf(deg_fwd[n], 1.0f);
    float inv_b = 1.0f / fmaxf(deg_bwd[n], 1.0f);

    size_t idx = (size_t)n * FEAT_D + (lane << 2);
    v4f vf = *(const v4f*)(sum_fwd + idx);
    v4f vb = *(const v4f*)(sum_bwd + idx);
#pragma unroll
    for (int j = 0; j < 4; ++j) { vf[j] *= inv_f; vb[j] *= inv_b; }
    *(v4f*)(sum_fwd + idx) = vf;
    *(v4f*)(sum_bwd + idx) = vb;
}

// ---------------------------------------------------------------------------
// WMMA GEMM: out = relu( [feat | fwd | bwd] @ W^T )
//   Block: 256 threads = 8 waves. Block owns 16 rows (M tile).
//   Wave w owns N tile [16w, 16w+16). Accumulate over K = 384 in steps of 4
//   with V_WMMA_F32_16X16X4_F32.
//
// Fragment layouts (ISA 7.12.2, wave32):
//   A (16x4, MxK):  lane l, vgpr r -> M = l&15,  K = 2*(l>>4) + r
//   B (4x16, KxN):  lane l, vgpr r -> N = l&15,  K = 2*(l>>4) + r
//   C/D (16x16):    lane l, vgpr r -> M = r + 8*(l>>4), N = l&15
// ---------------------------------------------------------------------------
__global__ __launch_bounds__(256) void wmma_gemm_relu_kernel(const float* __restrict__ feat,
                                                             const float* __restrict__ fwd,
                                                             const float* __restrict__ bwd,
                                                             const float* __restrict__ W,
                                                             float* __restrict__ out,
                                                             int n_nodes) {
    const int lane = threadIdx.x & 31;
    const int wave = threadIdx.x >> 5;          // 0..7 -> N tile
    const int m0   = blockIdx.x * 16;
    const int n0   = wave * 16;

    const int lidx = lane & 15;                 // M for A, N for B
    const int koff = (lane >> 4) * 2;           // 0 or 2

    // clamp (not mask): keeps EXEC all-ones around WMMA as the ISA requires
    int arow = m0 + lidx;
    if (arow >= n_nodes) arow = n_nodes - 1;

    const float* __restrict__ a_feat = feat + (size_t)arow * FEAT_D;
    const float* __restrict__ a_fwd  = fwd  + (size_t)arow * FEAT_D;
    const float* __restrict__ a_bwd  = bwd  + (size_t)arow * FEAT_D;
    const float* __restrict__ b_row  = W + (size_t)(n0 + lidx) * HID_D; // W[n, :]

    v8f acc = {};

    const float* asegs[3] = {a_feat, a_fwd, a_bwd};
#pragma unroll
    for (int seg = 0; seg < 3; ++seg) {
        const float* __restrict__ aseg = asegs[seg];
        const int kbase = seg * FEAT_D;
#pragma unroll 8
        for (int k = 0; k < FEAT_D; k += 4) {
            // (k + koff) is even -> 8-byte aligned b64 loads
            v2f a = *(const v2f*)(aseg + k + koff);
            v2f b = *(const v2f*)(b_row + kbase + k + koff);
            acc = __builtin_amdgcn_wmma_f32_16x16x4_f32(
                /*neg_a=*/false, a, /*neg_b=*/false, b,
                /*c_mod=*/(short)0, acc, /*reuse_a=*/false, /*reuse_b=*/false);
        }
    }

    // Store C/D with fused ReLU
    const int cm = (lane >> 4) * 8;
    const int cn = n0 + lidx;
#pragma unroll
    for (int r = 0; r < 8; ++r) {
        int row = m0 + cm + r;
        if (row < n_nodes) {
            float v = acc[r];
            out[(size_t)row * FEAT_D + cn] = v > 0.0f ? v : 0.0f;
        }
    }
}

// ---------------------------------------------------------------------------
extern "C" void kernel_launch(void* const* d_in, const int* in_sizes, int n_in,
                              void* d_out, int out_size, void* d_ws, size_t ws_size,
                              hipStream_t stream) {
    const float* feat = (const float*)d_in[0];      // [N, 128]
    const float* W    = (const float*)d_in[1];      // [128, 384]
    const int*   ei   = (const int*)d_in[2];        // [2, E]
    float* out = (float*)d_out;                     // [N, 128]

    const int n_nodes = in_sizes[0] / FEAT_D;
    const int n_edges = in_sizes[2] / 2;
    const int* src = ei;
    const int* dst = ei + n_edges;

    // workspace layout (floats): sum_fwd[N*128] | sum_bwd[N*128] | deg_fwd[N] | deg_bwd[N]
    float* sum_fwd = (float*)d_ws;
    float* sum_bwd = sum_fwd + (size_t)n_nodes * FEAT_D;
    float* deg_fwd = sum_bwd + (size_t)n_nodes * FEAT_D;
    float* deg_bwd = deg_fwd + n_nodes;

    const size_t zcount = (size_t)n_nodes * (2 * FEAT_D + 2);
    {
        size_t nvec = zcount >> 2;
        int blocks = (int)((nvec + 255) / 256);
        if (blocks > 262144) blocks = 262144;
        zero_ws_kernel<<<blocks, 256, 0, stream>>>((float*)d_ws, zcount);
    }

    {
        long long total = (long long)n_edges * 32;   // one wave per edge
        int blocks = (int)((total + 255) / 256);
        edge_scatter_kernel<<<blocks, 256, 0, stream>>>(src, dst, feat, sum_fwd, sum_bwd,
                                                        deg_fwd, deg_bwd, n_edges);
    }

    {
        long long total = (long long)n_nodes * 32;   // one wave per node
        int blocks = (int)((total + 255) / 256);
        finalize_kernel<<<blocks, 256, 0, stream>>>(sum_fwd, sum_bwd, deg_fwd, deg_bwd, n_nodes);
    }

    {
        int blocks = (n_nodes + 15) / 16;
        wmma_gemm_relu_kernel<<<blocks, 256, 0, stream>>>(feat, sum_fwd, sum_bwd, W, out, n_nodes);
    }
}